// LinearUpsampleBlock_3444563772233
// MI455X (gfx1250) — compile-verified
//
#include <hip/hip_runtime.h>

// LinearUpsampleBlock for MI455X (gfx1250).
//
// out[m, :] = sum_k w[m,k] * x[inds[m,k], :],  w = normalized 1/(d+1e-8)
//   x:     (262144, 128) f32  -> 128 MB, fits in 192 MB L2 (keep resident!)
//   inds:  (1048576, 3) i32
//   dists: (1048576, 3) f32
//   out:   (1048576, 128) f32 -> 512 MB streamed once (non-temporal stores)
//
// Pure memory-bound gather (HBM floor ~664 MB / 23.3 TB/s ~ 28 us).
// WMMA is inapplicable: every output row gathers its own 3x128 operand, so
// there is no shared B matrix for any v_wmma shape. The optimal kernel is a
// memory-path kernel: 4 rows per wave (lane = one float4 of each 128-ch row),
// scalar (SMEM) metadata loads, 12 saddr-form global_load_b128 gathers in
// flight per wave, and non-temporal b128 stores so the output stream does not
// evict x from L2.
//
// Weights are computed with a single division per row via
//   w_k = prod_{j!=k} d'_j / (d'_0 d'_1 + d'_1 d'_2 + d'_0 d'_2),  d' = d+1e-8
// which equals q_k / sum(q) exactly (up to rounding).

typedef float v4f __attribute__((ext_vector_type(4)));

namespace {
constexpr int kC           = 128;
constexpr int kVec         = kC / 4;                       // 32 float4 / row == 1 per lane
constexpr int kRowsPerWave = 4;
constexpr int kWavesPerBlk = 8;                            // 256 threads
constexpr int kRowsPerBlk  = kRowsPerWave * kWavesPerBlk;  // 32 rows / block
}

__device__ __forceinline__ void row_weights(const float* __restrict__ dists,
                                            int r, float w[3]) {
    const float d0 = dists[r * 3 + 0] + 1e-8f;
    const float d1 = dists[r * 3 + 1] + 1e-8f;
    const float d2 = dists[r * 3 + 2] + 1e-8f;
    const float p01 = d0 * d1, p12 = d1 * d2, p02 = d0 * d2;
    const float inv = 1.0f / (p01 + p12 + p02);   // the only division
    w[0] = p12 * inv;
    w[1] = p02 * inv;
    w[2] = p01 * inv;
}

__global__ __launch_bounds__(kWavesPerBlk * 32)
void linear_upsample_kernel(const float* __restrict__ x,
                            const int*   __restrict__ inds,
                            const float* __restrict__ dists,
                            float*       __restrict__ out,
                            int M)
{
    const int lane = threadIdx.x & 31;
    const int wave = threadIdx.x >> 5;

    // Wave-uniform base row; readfirstlane forces scalar-ness so metadata
    // loads become s_load_* (SMEM/KMcnt), and gather addresses are
    // SGPR-base + lane*16 (saddr-form global_load_b128).
    const int rowBase = __builtin_amdgcn_readfirstlane(
        (int)blockIdx.x * kRowsPerBlk + wave * kRowsPerWave);
    if (rowBase >= M) return;

    const v4f* __restrict__ x4   = (const v4f*)x;
    v4f*       __restrict__ out4 = (v4f*)out;

    if (rowBase + kRowsPerWave <= M) {
        // ---- Full-tile fast path: no per-row predicates (always taken for
        // ---- M = 1M since kRowsPerBlk divides it exactly).
        int   idx[kRowsPerWave][3];
        float w  [kRowsPerWave][3];

        // Phase 1: scalar metadata + weights (1 divide per row).
#pragma unroll
        for (int j = 0; j < kRowsPerWave; ++j) {
            const int r = rowBase + j;
            idx[j][0] = inds[r * 3 + 0];
            idx[j][1] = inds[r * 3 + 1];
            idx[j][2] = inds[r * 3 + 2];
            row_weights(dists, r, w[j]);
        }

        // Phase 2: issue all 12 b128 gathers for max memory-level parallelism;
        // x is L2-resident so these are mostly L2 hits.
        v4f g[kRowsPerWave][3];
#pragma unroll
        for (int j = 0; j < kRowsPerWave; ++j) {
#pragma unroll
            for (int k = 0; k < 3; ++k) {
                g[j][k] = x4[(size_t)idx[j][k] * kVec + lane];
            }
        }

        // Phase 3: combine + NON-TEMPORAL b128 stores (don't evict x from L2).
#pragma unroll
        for (int j = 0; j < kRowsPerWave; ++j) {
            v4f o = g[j][0] * w[j][0];
            o = o + g[j][1] * w[j][1];
            o = o + g[j][2] * w[j][2];
            __builtin_nontemporal_store(o, &out4[(size_t)(rowBase + j) * kVec + lane]);
        }
    } else {
        // ---- Tail path (never taken for the reference shapes).
        for (int r = rowBase; r < M; ++r) {
            float w[3];
            row_weights(dists, r, w);
            const int i0 = inds[r * 3 + 0];
            const int i1 = inds[r * 3 + 1];
            const int i2 = inds[r * 3 + 2];
            v4f o = x4[(size_t)i0 * kVec + lane] * w[0];
            o = o + x4[(size_t)i1 * kVec + lane] * w[1];
            o = o + x4[(size_t)i2 * kVec + lane] * w[2];
            __builtin_nontemporal_store(o, &out4[(size_t)r * kVec + lane]);
        }
    }
}

extern "C" void kernel_launch(void* const* d_in, const int* in_sizes, int n_in,
                              void* d_out, int out_size, void* d_ws, size_t ws_size,
                              hipStream_t stream)
{
    const float* x     = (const float*)d_in[0];
    const int*   inds  = (const int*)d_in[1];
    const float* dists = (const float*)d_in[2];
    float*       out   = (float*)d_out;

    const int M = in_sizes[1] / 3;  // upsample_inds is (M, 3)
    const int grid = (M + kRowsPerBlk - 1) / kRowsPerBlk;
    linear_upsample_kernel<<<grid, kWavesPerBlk * 32, 0, stream>>>(x, inds, dists, out, M);
}